// LiZAttention_6932077215993
// MI455X (gfx1250) — compile-verified
//
#include <hip/hip_runtime.h>
#include <math.h>

// Problem constants (fixed by the reference setup)
#define BH_TOT 32   // B*H = 2*16
#define NCHUNK 32   // NC
#define CT     128  // CHUNK * N_ORDERS
#define CPAIR  64   // CHUNK
#define DD     128  // D
#define STABV  10000.0f

typedef float v2f __attribute__((ext_vector_type(2)));
typedef float v8f __attribute__((ext_vector_type(8)));

__device__ __forceinline__ float gelu_tanh(float x) {
    const float k0 = 0.7978845608028654f;  // sqrt(2/pi)
    const float k1 = 0.044715f;
    float t = tanhf(k0 * (x + k1 * x * x * x));
    return 0.5f * x * (1.0f + t);
}

__device__ __forceinline__ float clipstab(float v) {
    // clip to [-STAB, STAB]; v_max_num/v_min_num also quiet NaNs toward the bound
    return fminf(fmaxf(v, -STABV), STABV);
}

// ---------------------------------------------------------------------------
// Phase 1: sequential scan over chunks. Only the last position (c=63) of each
// chunk feeds the recurrence, so per chunk we need 2 matvecs + a rank-2 update
// + gelu_norm of the 128x128 state. One block per (b,h); h lives in LDS.
// Writes the state entering each chunk to hstates[], and the final state.
// ---------------------------------------------------------------------------
__global__ __launch_bounds__(256) void liz_scan(
    const float* __restrict__ Wt, const float* __restrict__ Ut,
    const float* __restrict__ h0, float* __restrict__ hstates,
    float* __restrict__ hfinal)
{
    __shared__ float h[DD * DD];          // 64 KB state
    __shared__ float w0r[DD], w1r[DD], u0r[DD], u1r[DD];
    __shared__ float uv0[DD], uv1[DD];
    __shared__ float psum[256];
    __shared__ float wwred[64];

    const int tid = threadIdx.x;
    const int bh  = blockIdx.x;

    // load h0 (coalesced float4)
    {
        const float4* src = (const float4*)(h0 + (size_t)bh * DD * DD);
        float4* hv = (float4*)h;
        for (int i = tid; i < DD * DD / 4; i += 256) hv[i] = src[i];
    }
    __syncthreads();

    for (int ch = 0; ch < NCHUNK; ++ch) {
        // store state entering this chunk for phase 2
        {
            float4* dst = (float4*)(hstates + (size_t)(bh * NCHUNK + ch) * DD * DD);
            const float4* hv = (const float4*)h;
            for (int i = tid; i < DD * DD / 4; i += 256) dst[i] = hv[i];
        }
        const float* wbase = Wt + (size_t)(bh * NCHUNK + ch) * CT * DD;
        const float* ubase = Ut + (size_t)(bh * NCHUNK + ch) * CT * DD;
        if (tid < DD) {
            w0r[tid] = wbase[(size_t)(CT - 2) * DD + tid];
            w1r[tid] = wbase[(size_t)(CT - 1) * DD + tid];
            u0r[tid] = ubase[(size_t)(CT - 2) * DD + tid];
            u1r[tid] = ubase[(size_t)(CT - 1) * DD + tid];
        }
        __syncthreads();

        // kt0[d] = sum_r w0[r] * h[r][d]   (two half-sums per column)
        {
            int d = tid & (DD - 1);
            int half = tid >> 7;
            float s = 0.f;
            for (int r = half * 64; r < half * 64 + 64; ++r) s += w0r[r] * h[r * DD + d];
            psum[tid] = s;
        }
        __syncthreads();
        if (tid < DD) uv0[tid] = u0r[tid] - (psum[tid] + psum[tid + 128]);
        if (tid < 64) wwred[tid] = w1r[tid] * w0r[tid] + w1r[tid + 64] * w0r[tid + 64];
        __syncthreads();
        for (int off = 32; off > 0; off >>= 1) {
            if (tid < off) wwred[tid] += wwred[tid + off];
            __syncthreads();
        }
        float ww = wwred[0];

        // kt1[d] = sum_r w1[r]*h[r][d] + ww*uv0[d]
        {
            int d = tid & (DD - 1);
            int half = tid >> 7;
            float s = 0.f;
            for (int r = half * 64; r < half * 64 + 64; ++r) s += w1r[r] * h[r * DD + d];
            psum[tid] = s;
        }
        __syncthreads();
        if (tid < DD) uv1[tid] = u1r[tid] - (psum[tid] + psum[tid + 128] + ww * uv0[tid]);
        __syncthreads();

        // h += w0 (x) uv0 ; clip ; += w1 (x) uv1 ; clip ; accumulate row sq-sums
        {
            int row  = tid >> 1;
            int half = tid & 1;
            int base = row * DD + half * 64;
            float w0v = w0r[row], w1v = w1r[row];
            float sq = 0.f;
            for (int j = 0; j < 64; ++j) {
                int col = half * 64 + j;
                float v = clipstab(h[base + j] + w0v * uv0[col]);
                v = clipstab(v + w1v * uv1[col]);
                h[base + j] = v;
                sq += v * v;
            }
            psum[tid] = sq;
        }
        __syncthreads();
        // gelu_norm per row: n = ||row|| + 1e-6 ; row = n/2 * gelu(2*row/n)
        {
            int row  = tid >> 1;
            int half = tid & 1;
            float n  = sqrtf(psum[2 * row] + psum[2 * row + 1]) + 1e-6f;
            float inv = 2.0f / n;
            float hn  = 0.5f * n;
            int base = row * DD + half * 64;
            for (int j = 0; j < 64; ++j) {
                float x = h[base + j] * inv;
                h[base + j] = hn * gelu_tanh(x);
            }
        }
        __syncthreads();
    }

    {
        float4* dst = (float4*)(hfinal + (size_t)bh * DD * DD);
        const float4* hv = (const float4*)h;
        for (int i = tid; i < DD * DD / 4; i += 256) dst[i] = hv[i];
    }
}

// ---------------------------------------------------------------------------
// Phase 2: fully parallel outputs. One block per (bh, chunk): 8 waves.
// WH = W_chunk @ h and QH = Q_chunk @ h via V_WMMA_F32_16X16X4_F32
// (A = chunk rows striped per lane, B = h columns from LDS; each B-frag feeds
// two WMMAs). Epilogue fused in C-fragment registers:
//   even row t=2c:  uv0 = U - WH ; out = Q*uv0 + QH
//   odd  row t=2c+1: uv1 = U - WH - s_w[c]*uv0 ; out = Q*uv1 + QH + s_q[c]*uv0
// ---------------------------------------------------------------------------
__global__ __launch_bounds__(256) void liz_out(
    const float* __restrict__ Qt, const float* __restrict__ Wt,
    const float* __restrict__ Ut, const float* __restrict__ hstates,
    float* __restrict__ ys)
{
    __shared__ float h[DD * DD];  // 64 KB, B-matrix
    __shared__ float s_w[CPAIR], s_q[CPAIR];

    const int tid  = threadIdx.x;
    const int lane = tid & 31;
    const int wave = tid >> 5;
    const int nc = blockIdx.x, bh = blockIdx.y;

    const size_t cbase = (size_t)(bh * NCHUNK + nc) * CT * DD;
    const float* Qc = Qt + cbase;
    const float* Wc = Wt + cbase;
    const float* Uc = Ut + cbase;

    // load h state for this chunk (coalesced)
    {
        const float4* src = (const float4*)(hstates + (size_t)(bh * NCHUNK + nc) * DD * DD);
        float4* hv = (float4*)h;
        for (int i = tid; i < DD * DD / 4; i += 256) hv[i] = src[i];
    }

    // per-row-pair scalars: s_w[c] = w1.w0, s_q[c] = q1.w0  (wave-level dots)
    for (int i = 0; i < 8; ++i) {
        int c = wave * 8 + i;
        const float* w0p = Wc + (size_t)(2 * c) * DD;
        const float* w1p = Wc + (size_t)(2 * c + 1) * DD;
        const float* q1p = Qc + (size_t)(2 * c + 1) * DD;
        float pw = 0.f, pq = 0.f;
        for (int e = lane; e < DD; e += 32) {
            float a = w0p[e];
            pw += w1p[e] * a;
            pq += q1p[e] * a;
        }
        for (int off = 16; off > 0; off >>= 1) {
            pw += __shfl_xor(pw, off, 32);
            pq += __shfl_xor(pq, off, 32);
        }
        if (lane == 0) { s_w[c] = pw; s_q[c] = pq; }
    }
    __syncthreads();

    // A-fragment layout (16x4 f32): lane L -> row m*16 + L%16, K pair 2*(L/16)+{0,1}
    const int m      = wave;              // M-tile index (rows 16m..16m+15)
    const int mrow   = m * 16 + (lane & 15);
    const int khalf  = 2 * (lane >> 4);   // 0 or 2
    const int nlane  = lane & 15;
    const int rowoff = (lane >> 4) * 8;   // C-frag local-row offset

    // hoist both A row-strips (whole K=128) into VGPRs, reused for all 8 N-tiles
    v2f aW[32], aQ[32];
    {
        const float* wrow = Wc + (size_t)mrow * DD;
        const float* qrow = Qc + (size_t)mrow * DD;
#pragma unroll
        for (int k = 0; k < 32; ++k) {
            int col = 4 * k + khalf;
            aW[k] = *(const v2f*)(wrow + col);
            aQ[k] = *(const v2f*)(qrow + col);
        }
    }

    float* outbase = ys + cbase;
    for (int n = 0; n < 8; ++n) {
        v8f accW = {};
        v8f accQ = {};
        const int ncol = n * 16 + nlane;
#pragma unroll
        for (int k = 0; k < 32; ++k) {
            int kk = 4 * k + khalf;
            v2f b;
            b.x = h[kk * DD + ncol];
            b.y = h[(kk + 1) * DD + ncol];
            accW = __builtin_amdgcn_wmma_f32_16x16x4_f32(
                false, aW[k], false, b, (short)0, accW, false, false);
            accQ = __builtin_amdgcn_wmma_f32_16x16x4_f32(
                false, aQ[k], false, b, (short)0, accQ, false, false);
        }
        // fused epilogue: C-frag vgpr v -> local row v + rowoff, col = ncol
#pragma unroll
        for (int j = 0; j < 4; ++j) {
            int lr0 = 2 * j + rowoff;          // even local row (step 0)
            int t0  = m * 16 + lr0;
            int t1  = t0 + 1;
            int c   = t0 >> 1;
            float wh0 = accW[2 * j], wh1 = accW[2 * j + 1];
            float qh0 = accQ[2 * j], qh1 = accQ[2 * j + 1];
            float u0 = Uc[(size_t)t0 * DD + ncol];
            float u1 = Uc[(size_t)t1 * DD + ncol];
            float q0 = Qc[(size_t)t0 * DD + ncol];
            float q1 = Qc[(size_t)t1 * DD + ncol];
            float uv0v = u0 - wh0;
            float out0 = q0 * uv0v + qh0;
            float uv1v = u1 - (wh1 + s_w[c] * uv0v);
            float out1 = q1 * uv1v + qh1 + s_q[c] * uv0v;
            outbase[(size_t)t0 * DD + ncol] = out0;
            outbase[(size_t)t1 * DD + ncol] = out1;
        }
    }
}

extern "C" void kernel_launch(void* const* d_in, const int* in_sizes, int n_in,
                              void* d_out, int out_size, void* d_ws, size_t ws_size,
                              hipStream_t stream) {
    const float* q  = (const float*)d_in[0];  // (B,H,NC,CT,D)
    const float* W  = (const float*)d_in[1];  // (B,H,NC,CT,D)
    const float* U  = (const float*)d_in[2];  // (B,H,NC,CT,D)
    const float* h0 = (const float*)d_in[3];  // (B,H,D,D)

    float* ys     = (float*)d_out;                                  // (B,H,NC,CT,D)
    float* hfinal = ys + (size_t)BH_TOT * NCHUNK * CT * DD;         // (B,H,D,D)
    float* hstate = (float*)d_ws;   // BH*NC*D*D floats = 64 MB scratch

    liz_scan<<<BH_TOT, 256, 0, stream>>>(W, U, h0, hstate, hfinal);
    liz_out<<<dim3(NCHUNK, BH_TOT), 256, 0, stream>>>(q, W, U, hstate, ys);
}